// RotatedROICropper_86689619903163
// MI455X (gfx1250) — compile-verified
//
#include <hip/hip_runtime.h>

#define OUT_H   48
#define OUT_W   320
#define IMG_H   1024
#define IMG_W   1024
#define CHANS   3
#define BATCH   8
#define NBOX    64
#define NCROPS  (BATCH * NBOX)          // 512
#define QPR     (OUT_W / 4)             // 80 quads per row
#define QPC     (CHANS * OUT_H * QPR)   // 11520 quads per crop

typedef float vfloat4 __attribute__((ext_vector_type(4)));

// ---------------------------------------------------------------------------
// Kernel 1: per-box affine parameters (cv2 boxPoints -> affine map), 512 boxes.
// params[bn] = { p1x, p1y, dux, duy, dvx, dvy }
// sx = p1x + x*dux + y*dvx ; sy = p1y + x*duy + y*dvy
// theta is bounded to [-pi/4, pi/4] by the input distribution, so the
// hardware transcendentals (v_sin_f32 / v_cos_f32) are accurate here and
// avoid the Payne-Hanek reduction bloat of libm sinf/cosf.
// ---------------------------------------------------------------------------
__global__ void rroi_box_params(const float* __restrict__ boxes,
                                float* __restrict__ params) {
    int i = blockIdx.x * blockDim.x + threadIdx.x;
    if (i >= NCROPS) return;
    const float* b = boxes + (size_t)i * 5;
    float cx = b[0], cy = b[1], bw = b[2], bh = b[3], ang = b[4];
    float theta = -ang * 0.017453292519943295f;   // -deg * pi/180
    float s = __sinf(theta);                      // v_sin_f32 (TRANS32)
    float c = __cosf(theta);                      // v_cos_f32 (TRANS32)
    float a  = s * 0.5f;
    float bb = c * 0.5f;
    float p0x = cx - a * bh - bb * bw, p0y = cy + bb * bh - a * bw; // -> (0, H)
    float p1x = cx + a * bh - bb * bw, p1y = cy - bb * bh - a * bw; // -> (0, 0)
    float p2x = 2.0f * cx - p0x,       p2y = 2.0f * cy - p0y;      // -> (W, 0)
    float* o = params + (size_t)i * 6;
    o[0] = p1x;
    o[1] = p1y;
    o[2] = (p2x - p1x) * (1.0f / OUT_W);
    o[3] = (p2y - p1y) * (1.0f / OUT_W);
    o[4] = (p0x - p1x) * (1.0f / OUT_H);
    o[5] = (p0y - p1y) * (1.0f / OUT_H);
}

// ---------------------------------------------------------------------------
// Kernel 2: bilinear gather. One float4 of output per thread (4 consecutive x),
// branchless zero-padded 4-tap sampling, 128-bit coalesced NT stores.
// grid = (QPC/256 = 45, NCROPS = 512); block = 256 (8 waves, wave32).
//
// L2 policy: input (100 MB) + output (94 MB) vs 192 MB L2 -> keep the input
// resident (gathers have ~4x spatial reuse) and stream the write-once output
// past L2 with non-temporal stores.
// ---------------------------------------------------------------------------
__global__ __launch_bounds__(256)
void rroi_crop_kernel(const float* __restrict__ img,
                      const float* __restrict__ params,
                      float* __restrict__ out) {
    const int bn   = blockIdx.y;                        // crop index (b*NBOX+n)
    const int quad = blockIdx.x * 256 + threadIdx.x;    // 0 .. QPC-1
    const int c    = quad / (OUT_H * QPR);
    int rem        = quad - c * (OUT_H * QPR);
    const int y    = rem / QPR;
    const int qx   = (rem - y * QPR) * 4;               // first of 4 x coords

    // Block-uniform box params -> scalar loads.
    const float* p = params + (size_t)bn * 6;
    const float p1x = p[0], p1y = p[1];
    const float dux = p[2], duy = p[3];
    const float dvx = p[4], dvy = p[5];

    const float* __restrict__ src =
        img + ((size_t)(bn / NBOX) * CHANS + (size_t)c) * (size_t)(IMG_H * IMG_W);

    const float basex = fmaf((float)y, dvx, p1x);
    const float basey = fmaf((float)y, dvy, p1y);

    // gfx1250 prefetch: warm the cacheline(s) at the start of this thread's
    // source span before the dependent gather burst.
    {
        float psx = fmaf((float)qx, dux, basex);
        float psy = fmaf((float)qx, duy, basey);
        int px = min(max((int)psx, 0), IMG_W - 1);
        int py = min(max((int)psy, 0), IMG_H - 1);
        __builtin_prefetch(src + (size_t)py * IMG_W + px, 0, 3);
    }

    float res[4];
#pragma unroll
    for (int k = 0; k < 4; ++k) {
        const float fx = (float)(qx + k);
        const float sx = fmaf(fx, dux, basex);
        const float sy = fmaf(fx, duy, basey);
        const float x0f = floorf(sx);
        const float y0f = floorf(sy);
        const float dx = sx - x0f;
        const float dy = sy - y0f;
        const int x0 = (int)x0f;
        const int y0 = (int)y0f;
        const int x1 = x0 + 1;
        const int y1 = y0 + 1;

        // validity masks (zero padding outside image)
        const float vx0 = (x0 >= 0 && x0 < IMG_W) ? 1.0f : 0.0f;
        const float vx1 = (x1 >= 0 && x1 < IMG_W) ? 1.0f : 0.0f;
        const float vy0 = (y0 >= 0 && y0 < IMG_H) ? 1.0f : 0.0f;
        const float vy1 = (y1 >= 0 && y1 < IMG_H) ? 1.0f : 0.0f;

        const int x0c = min(max(x0, 0), IMG_W - 1);
        const int x1c = min(max(x1, 0), IMG_W - 1);
        const int y0c = min(max(y0, 0), IMG_H - 1);
        const int y1c = min(max(y1, 0), IMG_H - 1);

        const float* r0 = src + (size_t)y0c * IMG_W;
        const float* r1 = src + (size_t)y1c * IMG_W;
        const float i00 = r0[x0c];
        const float i10 = r0[x1c];
        const float i01 = r1[x0c];
        const float i11 = r1[x1c];

        const float w00 = (1.0f - dx) * (1.0f - dy) * (vx0 * vy0);
        const float w10 = dx * (1.0f - dy)          * (vx1 * vy0);
        const float w01 = (1.0f - dx) * dy          * (vx0 * vy1);
        const float w11 = dx * dy                   * (vx1 * vy1);

        res[k] = fmaf(i00, w00, fmaf(i10, w10, fmaf(i01, w01, i11 * w11)));
    }

    vfloat4 v = { res[0], res[1], res[2], res[3] };
    size_t oidx = (((size_t)bn * CHANS + (size_t)c) * OUT_H + (size_t)y) * OUT_W + (size_t)qx;
    // Non-temporal 128-bit store: stream output past L2 (write-once data).
    __builtin_nontemporal_store(v, (vfloat4*)(out + oidx));
}

// ---------------------------------------------------------------------------
extern "C" void kernel_launch(void* const* d_in, const int* in_sizes, int n_in,
                              void* d_out, int out_size, void* d_ws, size_t ws_size,
                              hipStream_t stream) {
    const float* x     = (const float*)d_in[0];   // [8,3,1024,1024] f32
    const float* boxes = (const float*)d_in[1];   // [8,64,5] f32
    float* out         = (float*)d_out;           // [512,3,48,320] f32
    float* params      = (float*)d_ws;            // 512 * 6 floats = 12 KB

    rroi_box_params<<<(NCROPS + 255) / 256, 256, 0, stream>>>(boxes, params);

    dim3 grid(QPC / 256, NCROPS, 1);              // (45, 512)
    rroi_crop_kernel<<<grid, 256, 0, stream>>>(x, params, out);
}